// HungarianMatcher_38293928411438
// MI455X (gfx1250) — compile-verified
//
#include <hip/hip_runtime.h>

typedef __attribute__((ext_vector_type(2))) float v2f;
typedef __attribute__((ext_vector_type(4))) float v4f;
typedef __attribute__((ext_vector_type(8))) float v8f;

#define NCOL 4096   // bs*T
#define LDS_STRIDE 132  // 128 cols + 4 pad floats: breaks 2-way bank conflict

// One 256-thread block = 8 waves; each wave owns one 16x16 WMMA tile.
// Block covers 16 rows x 128 cols of the output cost matrix.
__global__ __launch_bounds__(256) void hungarian_cost_kernel(
    const float* __restrict__ pred_logits,  // [8192][2]
    const float* __restrict__ pred_boxes,   // [8192][4]
    const float* __restrict__ tgt_pts,      // [4096][4]  (bs*T, npt*2)
    const int*   __restrict__ tgt_labels,   // [4096]
    float* __restrict__ out)                // [8192][4096]
{
    __shared__ float tile[16 * LDS_STRIDE];

    const int tid  = threadIdx.x;
    const int wave = tid >> 5;
    const int lane = tid & 31;
    const int half = lane >> 4;   // 0: lanes 0-15, 1: lanes 16-31
    const int l16  = lane & 15;

    const int rowBase = blockIdx.y * 16;
    const int colBlk  = blockIdx.x * 128;
    const int col     = colBlk + wave * 16 + l16;   // this lane's output column

    // ---- A fragment (16x4 f32): K=0,1 carry -sigmoid(logits); K=2,3 zero pad.
    // Layout: lanes 0-15 hold {K=0,K=1} for row M=lane; lanes 16-31 hold {K=2,K=3}.
    v2f a;
    if (half == 0) {
        const float* lg = pred_logits + (size_t)(rowBase + l16) * 2;
        float p0 = 1.0f / (1.0f + __expf(-lg[0]));
        float p1 = 1.0f / (1.0f + __expf(-lg[1]));
        a.x = -p0;
        a.y = -p1;
    } else {
        a.x = 0.0f;
        a.y = 0.0f;
    }

    // ---- B fragment (4x16 f32): rows K=0,1 = one-hot(tgt_label); K=2,3 zero.
    v2f b;
    {
        int id = tgt_labels[col];
        if (half == 0) {
            b.x = (id == 0) ? 1.0f : 0.0f;
            b.y = (id == 1) ? 1.0f : 0.0f;
        } else {
            b.x = 0.0f;
            b.y = 0.0f;
        }
    }

    // ---- cost_class tile via WMMA: D = A x B + 0  -> -prob[row, tgt_id[col]]
    v8f c = {};
    c = __builtin_amdgcn_wmma_f32_16x16x4_f32(
        /*neg_a=*/false, a, /*neg_b=*/false, b,
        /*c_mod=*/(short)0, c, /*reuse_a=*/false, /*reuse_b=*/false);

    // ---- L1 term. Target coords are column-constant across the 8 accum rows.
    v4f t = *(const v4f*)(tgt_pts + (size_t)col * 4);

#pragma unroll
    for (int v = 0; v < 8; ++v) {
        const int m = v + half * 8;  // row within 16-row tile (WMMA D layout)
        v4f bx = *(const v4f*)(pred_boxes + (size_t)(rowBase + m) * 4);
        float l1 = __builtin_fabsf(bx.x - t.x) + __builtin_fabsf(bx.y - t.y)
                 + __builtin_fabsf(bx.z - t.z) + __builtin_fabsf(bx.w - t.w);
        tile[m * LDS_STRIDE + wave * 16 + l16] = c[v] + l1;
    }

    __syncthreads();

    // ---- Coalesced writeback: each thread streams 8 contiguous floats of one
    // row (2 x float4) -> each output row segment is 512B fully-contiguous.
    const int r  = tid >> 4;        // 0..15
    const int cc = (tid & 15) * 8;  // 0,8,...,120
    v4f o0 = *(const v4f*)&tile[r * LDS_STRIDE + cc];
    v4f o1 = *(const v4f*)&tile[r * LDS_STRIDE + cc + 4];
    float* op = out + (size_t)(rowBase + r) * NCOL + colBlk + cc;
    *(v4f*)(op)     = o0;
    *(v4f*)(op + 4) = o1;
}

extern "C" void kernel_launch(void* const* d_in, const int* in_sizes, int n_in,
                              void* d_out, int out_size, void* d_ws, size_t ws_size,
                              hipStream_t stream) {
    (void)in_sizes; (void)n_in; (void)d_ws; (void)ws_size; (void)out_size;

    const float* pred_logits = (const float*)d_in[0];  // [8,1024,2]
    const float* pred_boxes  = (const float*)d_in[1];  // [8,1024,4]
    const float* tgt_pts     = (const float*)d_in[2];  // [8,512,2,2]
    const int*   tgt_labels  = (const int*)d_in[3];    // [8,512]
    float* out = (float*)d_out;                        // [8,1024,4096]

    dim3 grid(4096 / 128, 8192 / 16, 1);  // 32 x 512 blocks
    dim3 block(256, 1, 1);                // 8 waves (wave32)
    hipLaunchKernelGGL(hungarian_cost_kernel, grid, block, 0, stream,
                       pred_logits, pred_boxes, tgt_pts, tgt_labels, out);
}